// TextDetectionLoss_64166811402734
// MI455X (gfx1250) — compile-verified
//
#include <hip/hip_runtime.h>
#include <hip/hip_bf16.h>

// ---------------------------------------------------------------------------
// TextDetectionLoss for MI455X (gfx1250, wave32).
//   K1: streaming dice+bce reduction over 2x16MB (async-to-LDS double buffer)
//   K2: 5120-sample box gather + iou/smoothL1/conf-bce reduction (1 block)
//   K3: finalize; WMMA (v_wmma_f32_16x16x4_f32) wave reduction of partials
// ---------------------------------------------------------------------------

typedef float v2f __attribute__((ext_vector_type(2)));
typedef float v8f __attribute__((ext_vector_type(8)));
typedef int v4i __attribute__((ext_vector_type(4)));

#define GAS __attribute__((address_space(1)))
#define LAS __attribute__((address_space(3)))

#if defined(__AMDGCN__) && __has_builtin(__builtin_amdgcn_global_load_async_to_lds_b128)
#define HAVE_ASYNC_LDS 1
#else
#define HAVE_ASYNC_LDS 0
#endif

#if defined(__AMDGCN__) && __has_builtin(__builtin_amdgcn_s_wait_asynccnt)
#define ASYNC_WAIT(n) __builtin_amdgcn_s_wait_asynccnt(n)
#else
#define ASYNC_WAIT(n) asm volatile("s_wait_asynccnt %0" ::"n"(n) : "memory")
#endif

__device__ __forceinline__ void sched_fence() {
#if defined(__AMDGCN__) && __has_builtin(__builtin_amdgcn_sched_barrier)
  __builtin_amdgcn_sched_barrier(0);
#endif
}

// ---------------------------------------------------------------------------
// Wave32 full reduction via WMMA f32 (exact, layout-permutation-robust).
// A = {lane value, 0} (2 VGPRs), B = ones -> D[m][n] = rowsum_m(A).
// Sum of D VGPRs 0..7: lane<16 covers rows 0..7, lane>=16 covers rows 8..15.
// Total = e@lane0 + e@lane16 = sum of all A entries = sum over 32 lanes.
// ---------------------------------------------------------------------------
__device__ __forceinline__ float wave_sum32(float v) {
#if defined(__AMDGCN__) && __has_builtin(__builtin_amdgcn_wmma_f32_16x16x4_f32)
  v2f a;
  a[0] = v;
  a[1] = 0.0f;
  v2f b;
  b[0] = 1.0f;
  b[1] = 1.0f;
  v8f c = {};
  v8f d = __builtin_amdgcn_wmma_f32_16x16x4_f32(
      /*neg_a=*/false, a, /*neg_b=*/false, b,
      /*c_mod=*/(short)0, c, /*reuse_a=*/false, /*reuse_b=*/false);
  float e = d[0] + d[1] + d[2] + d[3] + d[4] + d[5] + d[6] + d[7];
  return __shfl(e, 0, 32) + __shfl(e, 16, 32);
#else
  for (int off = 16; off > 0; off >>= 1) v += __shfl_down(v, off, 32);
  return __shfl(v, 0, 32);
#endif
}

__device__ __forceinline__ void accum_elt(float p, float t, float& sb,
                                          float& si, float& sp, float& st) {
  // t is exactly 0.0f or 1.0f; computing both logs matches jnp exactly.
  sb += t * logf(p) + (1.0f - t) * logf(1.0f - p);
  si += p * t;
  sp += p;
  st += t;
}

// ---------------------------------------------------------------------------
// K1: dice/bce streaming reduction.
// N = 16*512*512 = 4,194,304 f32 -> 1,048,576 float4.
// Grid: 1024 blocks x 256 threads; 4 chunks of 256 float4 per block.
// ---------------------------------------------------------------------------
__global__ __launch_bounds__(256) void tdl_map_reduce(
    const float* __restrict__ pmap, const float* __restrict__ tmap,
    float4* __restrict__ partials) {
  const int tid = threadIdx.x;
  const unsigned base = blockIdx.x * 1024u;  // float4 index
  const float4* p4 = (const float4*)pmap;
  const float4* t4 = (const float4*)tmap;

  float sb = 0.f, si = 0.f, sp = 0.f, st = 0.f;

#if HAVE_ASYNC_LDS
  __shared__ float4 bufP[2][256];
  __shared__ float4 bufT[2][256];
  // Prefetch chunk 0 (2 async ops per wave per chunk; ASYNCcnt completes
  // in-order, so "wait <= 2" releases the previous chunk).
  __builtin_amdgcn_global_load_async_to_lds_b128(
      (GAS v4i*)(p4 + base + tid), (LAS v4i*)&bufP[0][tid], 0, 0);
  __builtin_amdgcn_global_load_async_to_lds_b128(
      (GAS v4i*)(t4 + base + tid), (LAS v4i*)&bufT[0][tid], 0, 0);
#pragma unroll
  for (int k = 0; k < 4; ++k) {
    const int cur = k & 1;
    if (k < 3) {
      __builtin_amdgcn_global_load_async_to_lds_b128(
          (GAS v4i*)(p4 + base + (k + 1) * 256 + tid),
          (LAS v4i*)&bufP[cur ^ 1][tid], 0, 0);
      __builtin_amdgcn_global_load_async_to_lds_b128(
          (GAS v4i*)(t4 + base + (k + 1) * 256 + tid),
          (LAS v4i*)&bufT[cur ^ 1][tid], 0, 0);
      ASYNC_WAIT(2);
    } else {
      ASYNC_WAIT(0);
    }
    sched_fence();
    // Each thread reads only its own LDS slot -> no barrier needed.
    float4 pv = bufP[cur][tid];
    float4 tv = bufT[cur][tid];
    accum_elt(pv.x, tv.x, sb, si, sp, st);
    accum_elt(pv.y, tv.y, sb, si, sp, st);
    accum_elt(pv.z, tv.z, sb, si, sp, st);
    accum_elt(pv.w, tv.w, sb, si, sp, st);
  }
#else
#pragma unroll
  for (int k = 0; k < 4; ++k) {
    float4 pv = p4[base + k * 256 + tid];
    float4 tv = t4[base + k * 256 + tid];
    accum_elt(pv.x, tv.x, sb, si, sp, st);
    accum_elt(pv.y, tv.y, sb, si, sp, st);
    accum_elt(pv.z, tv.z, sb, si, sp, st);
    accum_elt(pv.w, tv.w, sb, si, sp, st);
  }
#endif

  __shared__ float4 red[256];
  float4 acc;
  acc.x = sb;
  acc.y = si;
  acc.z = sp;
  acc.w = st;
  red[tid] = acc;
  __syncthreads();
  for (int s = 128; s > 0; s >>= 1) {
    if (tid < s) {
      float4 a = red[tid];
      float4 b = red[tid + s];
      a.x += b.x;
      a.y += b.y;
      a.z += b.z;
      a.w += b.w;
      red[tid] = a;
    }
    __syncthreads();
  }
  if (tid == 0) partials[blockIdx.x] = red[0];
}

// ---------------------------------------------------------------------------
// K2: box sampling + iou/smoothL1/conf-bce sums. Single block of 256 threads,
// 5120 samples total (B=16, NB=64, 5 points each).
// ---------------------------------------------------------------------------
__global__ __launch_bounds__(256) void tdl_box_reduce(
    const float* __restrict__ confp, const float* __restrict__ bbox,
    const float* __restrict__ boxes_tgt, float* __restrict__ out3) {
  const int tid = threadIdx.x;
  const int HW = 512 * 512;
  float slog = 0.f, ssl1 = 0.f, sconf = 0.f;

  for (int s = tid; s < 5120; s += 256) {
    int b = s / 320;
    int r = s - b * 320;
    int nb = r / 5;
    int pt = r - nb * 5;

    const float* bt = boxes_tgt + ((b * 64) + nb) * 5;
    float tc = bt[0];
    float tx1 = bt[1], ty1 = bt[2], tx2 = bt[3], ty2 = bt[4];

    int x1 = (int)floorf(tx1 * 512.f);
    int y1 = (int)floorf(ty1 * 512.f);
    int x2 = (int)floorf(tx2 * 512.f);
    int y2 = (int)floorf(ty2 * 512.f);
    x1 = x1 < 0 ? 0 : (x1 > 511 ? 511 : x1);
    y1 = y1 < 0 ? 0 : (y1 > 511 ? 511 : y1);
    x2 = x2 < 0 ? 0 : (x2 > 511 ? 511 : x2);
    y2 = y2 < 0 ? 0 : (y2 > 511 ? 511 : y2);

    int xs, ys;
    if (pt == 0) {
      xs = (x1 + x2) >> 1;
      ys = (y1 + y2) >> 1;
    } else if (pt == 1) {
      xs = x1;
      ys = y1;
    } else if (pt == 2) {
      xs = x2;
      ys = y1;
    } else if (pt == 3) {
      xs = x1;
      ys = y2;
    } else {
      xs = x2;
      ys = y2;
    }
    int idx = ys * 512 + xs;

    long bb = (long)b * 4 * HW + idx;
    float p0 = bbox[bb];
    float p1 = bbox[bb + HW];
    float p2 = bbox[bb + 2 * HW];
    float p3 = bbox[bb + 3 * HW];
    float pc = confp[(long)b * HW + idx];

    float pa = (p2 - p0) * (p3 - p1);
    float ta = (tx2 - tx1) * (ty2 - ty1);
    float ix1 = fmaxf(p0, tx1), iy1 = fmaxf(p1, ty1);
    float ix2 = fminf(p2, tx2), iy2 = fminf(p3, ty2);
    float inter = fmaxf(ix2 - ix1, 0.f) * fmaxf(iy2 - iy1, 0.f);
    float iou = inter / (pa + ta - inter + 1e-6f);
    slog += logf(iou + 1e-6f);

    float d;
    d = fabsf(p0 - tx1);
    ssl1 += (d < 1.f) ? 0.5f * d * d : d - 0.5f;
    d = fabsf(p1 - ty1);
    ssl1 += (d < 1.f) ? 0.5f * d * d : d - 0.5f;
    d = fabsf(p2 - tx2);
    ssl1 += (d < 1.f) ? 0.5f * d * d : d - 0.5f;
    d = fabsf(p3 - ty2);
    ssl1 += (d < 1.f) ? 0.5f * d * d : d - 0.5f;

    sconf += tc * logf(pc) + (1.f - tc) * logf(1.f - pc);
  }

  __shared__ float r0[256], r1[256], r2[256];
  r0[tid] = slog;
  r1[tid] = ssl1;
  r2[tid] = sconf;
  __syncthreads();
  for (int s = 128; s > 0; s >>= 1) {
    if (tid < s) {
      r0[tid] += r0[tid + s];
      r1[tid] += r1[tid + s];
      r2[tid] += r2[tid + s];
    }
    __syncthreads();
  }
  if (tid == 0) {
    out3[0] = r0[0];
    out3[1] = r1[0];
    out3[2] = r2[0];
  }
}

// ---------------------------------------------------------------------------
// K3: finalize. One wave (32 threads, EXEC all ones -> WMMA legal).
// Reduces 1024 float4 block partials via v_wmma_f32_16x16x4_f32.
// ---------------------------------------------------------------------------
__global__ __launch_bounds__(32) void tdl_finalize(
    const float4* __restrict__ partials, const float* __restrict__ box3,
    float* __restrict__ out) {
  const int lane = threadIdx.x;
  float a0 = 0.f, a1 = 0.f, a2 = 0.f, a3 = 0.f;
  for (int i = lane; i < 1024; i += 32) {
    float4 v = partials[i];
    a0 += v.x;
    a1 += v.y;
    a2 += v.z;
    a3 += v.w;
  }
  float Sb = wave_sum32(a0);  // sum of bce terms
  float Si = wave_sum32(a1);  // sum p*t
  float Sp = wave_sum32(a2);  // sum p
  float St = wave_sum32(a3);  // sum t

  const float N1 = 4194304.0f;  // 16*512*512
  float bce = -Sb / N1;
  float dice = 1.0f - (2.0f * Si + 1.0f) / (Sp + St + 1.0f);
  float text_loss = bce + dice;

  float slog = box3[0];
  float ssl1 = box3[1];
  float sconf = box3[2];
  float box_loss = -(slog / 5120.0f) + 0.5f * (ssl1 / 20480.0f);
  float conf_loss = -(sconf / 5120.0f);

  float total = 1.0f * text_loss + 5.0f * box_loss + 0.5f * conf_loss;
  if (lane == 0) {
    out[0] = total;
    out[1] = text_loss;
    out[2] = box_loss;
    out[3] = conf_loss;
  }
}

extern "C" void kernel_launch(void* const* d_in, const int* in_sizes, int n_in,
                              void* d_out, int out_size, void* d_ws,
                              size_t ws_size, hipStream_t stream) {
  (void)in_sizes;
  (void)n_in;
  (void)out_size;
  (void)ws_size;
  const float* text_map_pred = (const float*)d_in[0];
  const float* confidence_pred = (const float*)d_in[1];
  const float* bbox_coords_pred = (const float*)d_in[2];
  const float* text_map_tgt = (const float*)d_in[3];
  const float* boxes_tgt = (const float*)d_in[4];
  float* out = (float*)d_out;

  float4* partials = (float4*)d_ws;              // 1024 float4 = 16 KB
  float* box3 = ((float*)d_ws) + 4096;           // 3 floats after partials

  tdl_map_reduce<<<1024, 256, 0, stream>>>(text_map_pred, text_map_tgt,
                                           partials);
  tdl_box_reduce<<<1, 256, 0, stream>>>(confidence_pred, bbox_coords_pred,
                                        boxes_tgt, box3);
  tdl_finalize<<<1, 32, 0, stream>>>(partials, box3, out);
}